// SegEncodeLoss_37280316129713
// MI455X (gfx1250) — compile-verified
//
#include <hip/hip_runtime.h>
#include <cstdint>
#include <cstddef>

// ---------------------------------------------------------------------------
// SegEncodeLoss on MI455X (gfx1250).
//   preds:   [32768, 19] f32
//   targets: [32, 1024, 1024] i32 labels in [0,19)
//   out:     scalar f32 = mean BCE-with-logits vs per-32x32-block presence.
// Memory-bound: 128 MB of targets streamed once (~5.5 us at 23.3 TB/s).
// CDNA5 path: global_load_async_to_lds_b128 double-buffered per wave.
// ---------------------------------------------------------------------------

#define NUM_CLASSES     19
#define BLOCKS_TOTAL    32768      // 32 * (1024/32) * (1024/32)
#define WAVES_PER_WG    8
#define THREADS_PER_WG  256
#define BLOCKS_PER_WAVE 8
#define NUM_WAVES       (BLOCKS_TOTAL / BLOCKS_PER_WAVE)          // 4096
#define NUM_WGS         (NUM_WAVES / WAVES_PER_WG)                // 512
#define BLOCK_BYTES     4096       // 32x32 int32

// ---- CDNA5 async global->LDS (ASYNCcnt-tracked) ---------------------------

__device__ __forceinline__ void async_load_b128(uint32_t lds_off, const void* gaddr) {
    // VDST = VGPR holding LDS byte offset, VADDR = 64-bit global address, GV mode.
    asm volatile("global_load_async_to_lds_b128 %0, %1, off"
                 :: "v"(lds_off), "v"(gaddr)
                 : "memory");
}

__device__ __forceinline__ void wait_async_le8() {
    asm volatile("s_wait_asynccnt 0x8" ::: "memory");
}
__device__ __forceinline__ void wait_async_0() {
    asm volatile("s_wait_asynccnt 0x0" ::: "memory");
}

// Issue 8 wave-wide async b128 loads covering one 32x32 int32 block.
// Lane L, step k: global row (4k + L/8), 16B chunk (L%8) of that row.
// LDS layout: contiguous 4 KB, byte (k*512 + L*16).
__device__ __forceinline__ void issue_block_async(const int* targets, int n,
                                                  uint32_t lds_base_off, int lane) {
    const int b  = n >> 10;          // batch
    const int by = (n >> 5) & 31;    // block row
    const int bx = n & 31;           // block col
    const char* gbase = (const char*)(targets
                        + (size_t)b * (1024u * 1024u)
                        + (size_t)by * (32u * 1024u)
                        + (size_t)bx * 32u);
    const char* glane = gbase + (size_t)(lane >> 3) * 4096u + (size_t)(lane & 7) * 16u;
    const uint32_t lbase = lds_base_off + (uint32_t)lane * 16u;
#pragma unroll
    for (int k = 0; k < 8; ++k) {
        async_load_b128(lbase + (uint32_t)k * 512u, glane + (size_t)k * 16384u);
    }
}

// OR together class bits of one block from LDS (lane reads 8 x int4, 512B stride).
__device__ __forceinline__ uint32_t block_presence(const char* lds_block, int lane) {
    const int4* p = (const int4*)(lds_block + lane * 16);
    uint32_t m = 0u;
#pragma unroll
    for (int k = 0; k < 8; ++k) {
        int4 v = p[k * 32];          // 32 * sizeof(int4) = 512 B
        m |= (1u << v.x) | (1u << v.y) | (1u << v.z) | (1u << v.w);
    }
    // wave32 OR-reduce; all lanes end with the full mask
#pragma unroll
    for (int off = 16; off >= 1; off >>= 1)
        m |= (uint32_t)__shfl_xor((int)m, off, 32);
    return m;
}

__global__ __launch_bounds__(THREADS_PER_WG)
void seg_encode_loss_kernel(const float* __restrict__ preds,
                            const int* __restrict__ targets,
                            float* __restrict__ partials) {
    __shared__ char smem[WAVES_PER_WG * 2 * BLOCK_BYTES];   // 64 KB / WG

    const int lane = threadIdx.x & 31;
    const int wv   = threadIdx.x >> 5;
    const int waveGlobal = blockIdx.x * WAVES_PER_WG + wv;
    const int n0 = waveGlobal * BLOCKS_PER_WAVE;

    char* buf[2] = { smem + (wv * 2 + 0) * BLOCK_BYTES,
                     smem + (wv * 2 + 1) * BLOCK_BYTES };
    // Flat LDS byte offsets for the async instruction: generic address low 32
    // bits == LDS offset (flat aperture mapping, ISA 10.2).
    uint32_t bufOff[2] = { (uint32_t)(size_t)buf[0], (uint32_t)(size_t)buf[1] };

    // Prime the pipeline.
    issue_block_async(targets, n0, bufOff[0], lane);

    float acc = 0.0f;
#pragma unroll
    for (int i = 0; i < BLOCKS_PER_WAVE; ++i) {
        const int cur = i & 1;
        if (i + 1 < BLOCKS_PER_WAVE) {
            issue_block_async(targets, n0 + i + 1, bufOff[(i + 1) & 1], lane);
            wait_async_le8();   // 8 newest may be in flight; block i has landed
        } else {
            wait_async_0();
        }

        const uint32_t m = block_presence(buf[cur], lane);

        // Stable BCE-with-logits: max(x,0) - x*t + log1p(exp(-|x|)); lanes 0..18.
        const int n = n0 + i;
        float e = 0.0f;
        if (lane < NUM_CLASSES) {
            const float x = preds[(size_t)n * NUM_CLASSES + lane];
            const float t = (float)((m >> lane) & 1u);
            e = fmaxf(x, 0.0f) - x * t + log1pf(expf(-fabsf(x)));
        }
#pragma unroll
        for (int off = 16; off >= 1; off >>= 1)
            e += __shfl_xor(e, off, 32);        // fixed tree -> deterministic
        acc += e;                                // same value in every lane
    }

    if (lane == 0) partials[waveGlobal] = acc;
}

__global__ __launch_bounds__(THREADS_PER_WG)
void seg_encode_loss_reduce(const float* __restrict__ partials,
                            float* __restrict__ out) {
    __shared__ float sb[THREADS_PER_WG];
    const int t = threadIdx.x;
    float s = 0.0f;
#pragma unroll
    for (int j = 0; j < NUM_WAVES / THREADS_PER_WG; ++j)   // 16, fixed order
        s += partials[t + j * THREADS_PER_WG];
    sb[t] = s;
    __syncthreads();
#pragma unroll
    for (int stride = THREADS_PER_WG / 2; stride >= 1; stride >>= 1) {
        if (t < stride) sb[t] += sb[t + stride];
        __syncthreads();
    }
    if (t == 0)
        out[0] = sb[0] * (1.0f / ((float)BLOCKS_TOTAL * (float)NUM_CLASSES));
}

extern "C" void kernel_launch(void* const* d_in, const int* in_sizes, int n_in,
                              void* d_out, int out_size, void* d_ws, size_t ws_size,
                              hipStream_t stream) {
    (void)in_sizes; (void)n_in; (void)out_size; (void)ws_size;
    const float* preds   = (const float*)d_in[0];
    const int*   targets = (const int*)d_in[1];
    // d_in[2] = grid_size (constant 32, baked into the kernel)

    float* partials = (float*)d_ws;            // 4096 floats = 16 KB
    float* out      = (float*)d_out;

    seg_encode_loss_kernel<<<NUM_WGS, THREADS_PER_WG, 0, stream>>>(preds, targets, partials);
    seg_encode_loss_reduce<<<1, THREADS_PER_WG, 0, stream>>>(partials, out);
}